// GCN3_57071525429592
// MI455X (gfx1250) — compile-verified
//
#include <hip/hip_runtime.h>

#define NN 100000
#define EE 1600000
#define HH 128
#define GG 512
#define CC 10

typedef float v2f __attribute__((ext_vector_type(2)));
typedef float v8f __attribute__((ext_vector_type(8)));

// ---------------- utility kernels ----------------

__global__ void k_fill(float* __restrict__ p, float v, size_t n) {
  size_t i = (size_t)blockIdx.x * blockDim.x + threadIdx.x;
  if (i < n) p[i] = v;
}

__global__ void k_deg_accum(const int* __restrict__ dst, float* __restrict__ deg, int ne) {
  int e = blockIdx.x * blockDim.x + threadIdx.x;
  if (e < ne) atomicAdd(&deg[dst[e]], 1.0f);
}

__global__ void k_rsqrt(float* __restrict__ d, int n) {
  int i = blockIdx.x * blockDim.x + threadIdx.x;
  if (i < n) d[i] = rsqrtf(fmaxf(d[i], 1.0f));
}

// ---------------- fp32 WMMA GEMM: C[n x 128] = A[n x 128] * W[128 x 128] ----------------
// block = 256 threads = 8 waves (wave32). Each block: 16 rows x 128 cols.
// A tile (16x128 = 8KB) staged once in LDS and shared by all 8 waves;
// wave w owns column tile col0 = 16*w; K looped in steps of 4 via V_WMMA_F32_16X16X4_F32.
__global__ __launch_bounds__(256) void k_gemm128(const float* __restrict__ A,
                                                 const float* __restrict__ W,
                                                 float* __restrict__ C) {
  __shared__ float As[16 * HH];  // 8 KB

  const int tid  = threadIdx.x;
  const int row0 = blockIdx.x * 16;

  // cooperative stage: 16x128 floats = 512 float4; 256 threads x 2
  {
    const float4* __restrict__ Ag = (const float4*)(A + (size_t)row0 * HH);
    float4* As4 = (float4*)As;
    As4[tid]       = Ag[tid];
    As4[tid + 256] = Ag[tid + 256];
  }
  __syncthreads();

  const int wave  = tid >> 5;
  const int lane  = tid & 31;
  const int lhalf = lane >> 4;      // 0: lanes 0-15, 1: lanes 16-31
  const int l15   = lane & 15;
  const int col0  = wave * 16;

  const float* Arow = As + l15 * HH;

  v8f acc = {};
#pragma unroll
  for (int kt = 0; kt < 32; ++kt) {
    // A 16x4 layout: VGPR0 = K {0 | 2}, VGPR1 = K {1 | 3} (low/high lane half)
    const int k = kt * 4 + lhalf * 2;
    v2f a = *(const v2f*)(Arow + k);          // ds_load_b64
    v2f b;
    // B 4x16 layout: VGPR0 = row K {0 | 2}, VGPR1 = row K {1 | 3}; N striped on lanes 0-15
    b.x = W[(size_t)k * HH + col0 + l15];
    b.y = W[(size_t)(k + 1) * HH + col0 + l15];
    acc = __builtin_amdgcn_wmma_f32_16x16x4_f32(false, a, false, b,
                                                (short)0, acc, false, false);
  }

  // D 16x16 f32: VGPR r -> M = r (lanes 0-15) or r+8 (lanes 16-31), N = lane&15
  float* __restrict__ Crow = C + (size_t)(row0 + lhalf * 8) * HH + col0 + l15;
#pragma unroll
  for (int r = 0; r < 8; ++r) Crow[(size_t)r * HH] = acc[r];
}

// ---------------- edge aggregation: out[dst] += norm * h[src] ----------------
// one wave per edge; each lane moves 4 consecutive floats (float4 gather, 4 atomics)
__global__ __launch_bounds__(256) void k_edge_agg(const float* __restrict__ h,
                                                  const int* __restrict__ src,
                                                  const int* __restrict__ dst,
                                                  const float* __restrict__ dis,
                                                  float* __restrict__ out, int ne) {
  const int e = blockIdx.x * 8 + (threadIdx.x >> 5);
  if (e >= ne) return;
  const int lane = threadIdx.x & 31;
  const int s = src[e];
  const int d = dst[e];
  const float nrm = dis[s] * dis[d];
  const float4 v = ((const float4*)(h + (size_t)s * HH))[lane];
  float* o = out + (size_t)d * HH + lane * 4;
  atomicAdd(o + 0, v.x * nrm);
  atomicAdd(o + 1, v.y * nrm);
  atomicAdd(o + 2, v.z * nrm);
  atomicAdd(o + 3, v.w * nrm);
}

// self-loop contribution (norm = dis^2 = 1/deg) plus bias
__global__ void k_self_bias(const float* __restrict__ h, const float* __restrict__ dis,
                            const float* __restrict__ b, float* __restrict__ out, int n) {
  size_t i = (size_t)blockIdx.x * blockDim.x + threadIdx.x;
  if (i >= (size_t)n * HH) return;
  const int node = (int)(i >> 7);
  const int f = (int)(i & (HH - 1));
  const float dn = dis[node];
  out[i] += dn * dn * h[i] + b[f];
}

// ---------------- batchnorm over relu(h): stats then in-place apply ----------------
__global__ __launch_bounds__(128) void k_stats(const float* __restrict__ h,
                                               float* __restrict__ stats, int n) {
  const int f = threadIdx.x;              // feature 0..127
  const int start = blockIdx.x * 256;
  const int end = min(start + 256, n);
  float s = 0.f, s2 = 0.f;
  for (int i = start; i < end; ++i) {
    float v = fmaxf(h[(size_t)i * HH + f], 0.f);
    s += v;
    s2 += v * v;
  }
  atomicAdd(&stats[f], s);
  atomicAdd(&stats[HH + f], s2);
}

__global__ void k_bn_apply(float* __restrict__ h, const float* __restrict__ stats,
                           const float* __restrict__ gamma, const float* __restrict__ beta,
                           int n) {
  size_t i = (size_t)blockIdx.x * blockDim.x + threadIdx.x;
  if (i >= (size_t)n * HH) return;
  const int f = (int)(i & (HH - 1));
  const float invN = 1.0f / (float)n;
  const float m = stats[f] * invN;
  const float var = stats[HH + f] * invN - m * m;
  const float v = fmaxf(h[i], 0.f);
  h[i] = (v - m) * rsqrtf(var + 1e-5f) * gamma[f] + beta[f];
}

// ---------------- pooling + classifier ----------------
__global__ void k_count(const int* __restrict__ batch, float* __restrict__ counts, int n) {
  int i = blockIdx.x * blockDim.x + threadIdx.x;
  if (i < n) atomicAdd(&counts[batch[i]], 1.0f);
}

__global__ void k_pool(const float* __restrict__ h, const int* __restrict__ batch,
                       float* __restrict__ pooled, int n) {
  size_t i = (size_t)blockIdx.x * blockDim.x + threadIdx.x;
  if (i >= (size_t)n * HH) return;
  const int node = (int)(i >> 7);
  const int f = (int)(i & (HH - 1));
  atomicAdd(&pooled[(size_t)batch[node] * HH + f], h[i]);
}

__global__ void k_final(const float* __restrict__ pooled, const float* __restrict__ counts,
                        const float* __restrict__ Wf, const float* __restrict__ bf,
                        float* __restrict__ out) {
  int idx = blockIdx.x * blockDim.x + threadIdx.x;
  if (idx >= GG * CC) return;
  const int g = idx / CC;
  const int c = idx % CC;
  const float inv = 1.0f / fmaxf(counts[g], 1.0f);
  float acc = 0.f;
#pragma unroll 4
  for (int f = 0; f < HH; ++f) acc += pooled[(size_t)g * HH + f] * Wf[f * CC + c];
  out[idx] = acc * inv + bf[c];
}

// ---------------- driver ----------------
extern "C" void kernel_launch(void* const* d_in, const int* in_sizes, int n_in,
                              void* d_out, int out_size, void* d_ws, size_t ws_size,
                              hipStream_t stream) {
  const float* x     = (const float*)d_in[0];
  const int*   ei    = (const int*)d_in[1];
  const int*   batch = (const int*)d_in[2];
  const float* W0 = (const float*)d_in[3];
  const float* b0 = (const float*)d_in[4];
  const float* W1 = (const float*)d_in[5];
  const float* b1 = (const float*)d_in[6];
  const float* W2 = (const float*)d_in[7];
  const float* b2 = (const float*)d_in[8];
  const float* g0 = (const float*)d_in[9];
  const float* be0 = (const float*)d_in[10];
  const float* g1 = (const float*)d_in[11];
  const float* be1 = (const float*)d_in[12];
  const float* Wf = (const float*)d_in[13];
  const float* bf = (const float*)d_in[14];
  float* out = (float*)d_out;

  const int* src = ei;
  const int* dst = ei + EE;

  float* ws     = (float*)d_ws;
  float* buf0   = ws;                              // N*H  (GEMM output h)
  float* buf1   = buf0 + (size_t)NN * HH;          // N*H  (aggregated / normalized)
  float* dis    = buf1 + (size_t)NN * HH;          // N
  float* stats  = dis + NN;                        // 2*H
  float* pooled = stats + 2 * HH;                  // G*H
  float* counts = pooled + (size_t)GG * HH;        // G

  const size_t nh = (size_t)NN * HH;
  const int TB = 256;
  const int gridNH = (int)((nh + TB - 1) / TB);

  // ---- degree / normalization ----
  k_fill<<<(NN + TB - 1) / TB, TB, 0, stream>>>(dis, 1.0f, NN);     // self-loop
  k_deg_accum<<<(EE + TB - 1) / TB, TB, 0, stream>>>(dst, dis, EE);
  k_rsqrt<<<(NN + TB - 1) / TB, TB, 0, stream>>>(dis, NN);

  const float* layer_in[3]   = {x, buf1, buf1};
  const float* layer_W[3]    = {W0, W1, W2};
  const float* layer_b[3]    = {b0, b1, b2};
  const float* layer_g[3]    = {g0, g1, nullptr};
  const float* layer_beta[3] = {be0, be1, nullptr};

  for (int l = 0; l < 3; ++l) {
    // h = in @ W  (WMMA fp32, LDS-staged A tile)
    k_gemm128<<<NN / 16, 256, 0, stream>>>(layer_in[l], layer_W[l], buf0);
    // aggregate into buf1
    k_fill<<<gridNH, TB, 0, stream>>>(buf1, 0.0f, nh);
    k_edge_agg<<<(EE + 7) / 8, 256, 0, stream>>>(buf0, src, dst, dis, buf1, EE);
    k_self_bias<<<gridNH, TB, 0, stream>>>(buf0, dis, layer_b[l], buf1, NN);
    if (l < 2) {
      k_fill<<<1, 256, 0, stream>>>(stats, 0.0f, 2 * HH);
      k_stats<<<(NN + 255) / 256, 128, 0, stream>>>(buf1, stats, NN);
      k_bn_apply<<<gridNH, TB, 0, stream>>>(buf1, stats, layer_g[l], layer_beta[l], NN);
    }
  }

  // ---- global mean pool + classifier ----
  k_fill<<<((size_t)GG * HH + TB - 1) / TB, TB, 0, stream>>>(pooled, 0.0f, (size_t)GG * HH);
  k_fill<<<1, 256, 0, stream>>>(counts, 0.0f, GG);
  k_count<<<(NN + TB - 1) / TB, TB, 0, stream>>>(batch, counts, NN);
  k_pool<<<gridNH, TB, 0, stream>>>(buf1, batch, pooled, NN);
  k_final<<<(GG * CC + TB - 1) / TB, TB, 0, stream>>>(pooled, counts, Wf, bf, out);
}